// Metamorph_parameterReinforcer_62096637165904
// MI455X (gfx1250) — compile-verified
//
#include <hip/hip_runtime.h>

typedef float v2f __attribute__((ext_vector_type(2)));
typedef float v8f __attribute__((ext_vector_type(8)));

#define MODES 4096
#define NL    2048
#define OUT2  (NL * 8)

// Re(tanh(a + i*b)) = sinh(2a) / (cosh(2a) + cos(2b)), robust for large |a|.
__device__ __forceinline__ float ctanh_re(float x, float sr, float si) {
    float a2 = 2.0f * x * sr;
    float b2 = 2.0f * x * si;
    if (fabsf(a2) > 30.0f) return copysignf(1.0f, a2);
    float e  = __expf(a2);
    float ei = 1.0f / e;
    float sh = 0.5f * (e - ei);
    float ch = 0.5f * (e + ei);
    return sh / (ch + __cosf(b2));
}

// One block (256 threads): reduce S_i = sum(w_fft_i) (complex), then
// x[f] = Re tanh(Re tanh(Re tanh(model_p[f,0]*S0)*S1)*S2), f = 0..2047.
__global__ void __launch_bounds__(256)
fft_feature_kernel(const float* __restrict__ model_p,
                   const float* __restrict__ w0,
                   const float* __restrict__ w1,
                   const float* __restrict__ w2,
                   float* __restrict__ x_out) {
    __shared__ float red[512];
    __shared__ float S[6];
    const int t = threadIdx.x;
    const float* wf[3] = {w0, w1, w2};
    for (int i = 0; i < 3; ++i) {
        float sr = 0.0f, si = 0.0f;
        for (int j = t; j < MODES; j += 256) {
            v2f c = *(const v2f*)(wf[i] + 2 * j);   // interleaved complex64
            sr += c.x; si += c.y;
        }
        red[t] = sr; red[256 + t] = si;
        __syncthreads();
        for (int ofs = 128; ofs > 0; ofs >>= 1) {
            if (t < ofs) { red[t] += red[t + ofs]; red[256 + t] += red[256 + t + ofs]; }
            __syncthreads();
        }
        if (t == 0) { S[2 * i] = red[0]; S[2 * i + 1] = red[256]; }
        __syncthreads();
    }
    const float s0r = S[0], s0i = S[1], s1r = S[2], s1i = S[3], s2r = S[4], s2i = S[5];
    for (int f = t; f < NL; f += 256) {
        float v = model_p[(size_t)f * MODES];       // first FFT column only
        v = ctanh_re(v, s0r, s0i);
        v = ctanh_re(v, s1r, s1i);
        v = ctanh_re(v, s2r, s2i);
        x_out[f] = v;
    }
}

// GEMV via V_WMMA_F32_16X16X4_F32.
// Block = 256 threads = 8 waves, owns 16 output rows (row0 = blockIdx.x*16).
// Wave w covers the K-slice [w*K/8, (w+1)*K/8).
// A = 16x4 f32 weight tile: lane layout (M=lane%16, K pair = 2*(lane/16)),
// i.e. one contiguous float2 per lane -> global_load_b64.
// B = vec segment replicated across all N columns: same per-lane K pair,
// one float2 per lane -> ds_load_b64 from the LDS-staged vector.
// Inner loop is chunked by 32 K-elements: 8 load pairs issued before the
// 8-deep WMMA accumulation chain, so 8 loadcnt entries stay in flight.
// C/D layout: column N=0 lives in lane 0 (M=0..7 in c[0..7]) and lane 16
// (M=8..15 in c[0..7]); partials are reduced across the 8 waves in LDS.
template<int K, bool DO_TANH>
__global__ void __launch_bounds__(256)
gemv16_wmma(const float* __restrict__ W, const float* __restrict__ vec,
            const float* __restrict__ bias, float* __restrict__ out) {
    __shared__ float smem[K + 128];
    float* sv   = smem;       // K floats: the input vector
    float* part = smem + K;   // 8 waves * 16 rows of partial dots

    const int t = threadIdx.x;
#pragma unroll
    for (int j = t * 4; j < K; j += 256 * 4)
        *(float4*)(sv + j) = *(const float4*)(vec + j);
    __syncthreads();

    const int lane  = t & 31;
    const int wave  = t >> 5;
    const int row0  = blockIdx.x * 16;
    const int r     = lane & 15;
    const int khalf = (lane >> 4) << 1;               // 0 or 2 (K sub-pair)

    constexpr int KPW = K / 8;                        // K elems per wave
    const float* wp = W + (size_t)(row0 + r) * K + wave * KPW + khalf;
    const float* vp = sv + wave * KPW + khalf;

    v8f c = {};
#pragma unroll 1
    for (int k = 0; k < KPW; k += 32) {               // 8 WMMA steps / chunk
        v2f a[8], b[8];
#pragma unroll
        for (int u = 0; u < 8; ++u) {
            a[u] = *(const v2f*)(wp + k + 4 * u);     // global_load_b64 x8
            b[u] = *(const v2f*)(vp + k + 4 * u);     // ds_load_b64 x8
        }
#pragma unroll
        for (int u = 0; u < 8; ++u)
            c = __builtin_amdgcn_wmma_f32_16x16x4_f32(
                    false, a[u], false, b[u], (short)0, c, false, false);
    }

    if (lane == 0) {
#pragma unroll
        for (int i = 0; i < 8; ++i) part[wave * 16 + i] = c[i];
    }
    if (lane == 16) {
#pragma unroll
        for (int i = 0; i < 8; ++i) part[wave * 16 + 8 + i] = c[i];
    }
    __syncthreads();

    if (t < 16) {
        float acc = bias[row0 + t];
#pragma unroll
        for (int w = 0; w < 8; ++w) acc += part[w * 16 + t];
        out[row0 + t] = DO_TANH ? tanhf(acc) : acc;
    }
}

extern "C" void kernel_launch(void* const* d_in, const int* in_sizes, int n_in,
                              void* d_out, int out_size, void* d_ws, size_t ws_size,
                              hipStream_t stream) {
    const float* model_p = (const float*)d_in[0];
    const float* w0      = (const float*)d_in[1];
    const float* w1      = (const float*)d_in[2];
    const float* w2      = (const float*)d_in[3];
    const float* lin1_w  = (const float*)d_in[4];
    const float* lin1_b  = (const float*)d_in[5];
    const float* lin2_w  = (const float*)d_in[6];
    const float* lin2_b  = (const float*)d_in[7];
    float* out = (float*)d_out;

    float* x = (float*)d_ws;     // 2048 floats
    float* h = x + NL;           // 4096 floats

    fft_feature_kernel<<<1, 256, 0, stream>>>(model_p, w0, w1, w2, x);

    // h = tanh(lin1_w(4096x2048) @ x + b1)
    gemv16_wmma<NL, true><<<MODES / 16, 256, 0, stream>>>(lin1_w, x, lin1_b, h);

    // out = lin2_w(16384x4096) @ h + b2
    gemv16_wmma<MODES, false><<<OUT2 / 16, 256, 0, stream>>>(lin2_w, h, lin2_b, out);
}